// MyOutputPPBlock_55405078118414
// MI455X (gfx1250) — compile-verified
//
#include <hip/hip_runtime.h>
#include <hip/hip_bf16.h>

#define HIDDEN   128
#define OUT_EMB  256
#define NNODES   50000

typedef float v2f __attribute__((ext_vector_type(2)));
typedef float v8f __attribute__((ext_vector_type(8)));

// ---------------------------------------------------------------------------
// Zero-fill (agg accumulator must start at 0; harness poisons ws with 0xAA)
// ---------------------------------------------------------------------------
__global__ __launch_bounds__(256) void zero_kernel(float* __restrict__ p, int n4) {
    int t = blockIdx.x * 256 + threadIdx.x;
    if (t < n4) ((float4*)p)[t] = make_float4(0.f, 0.f, 0.f, 0.f);
}

// ---------------------------------------------------------------------------
// Edge phase: h[e,k] = (sum_r rbf[e,r] * W_rbf[k,r]) * x[e,k]
//             agg[i[e],k] += h[e,k]      (fp32 atomics, agg is L2-resident)
// One wave (32 lanes) per edge; each lane owns 4 hidden channels via a
// float4 load of x (global_load_b128, row fully coalesced per wave).
// W_rbf (3KB) and the block's 8 rbf rows staged in LDS.
// ---------------------------------------------------------------------------
__global__ __launch_bounds__(256)
void edge_scatter_kernel(const float* __restrict__ x,    // [E,128]
                         const float* __restrict__ rbf,  // [E,6]
                         const int*   __restrict__ idx,  // [E]
                         const float* __restrict__ Wrbf, // [128,6]
                         float* __restrict__ agg,        // [NNODES,128]
                         int E)
{
    __shared__ float w[HIDDEN * 6];   // W_rbf
    __shared__ float rb[8 * 6];       // rbf rows for this block's 8 edges

    const int ebase = blockIdx.x * 8;
    for (int t = threadIdx.x; t < HIDDEN * 6; t += 256) w[t] = Wrbf[t];
    if (threadIdx.x < 48) {
        int e = ebase + (threadIdx.x / 6);
        int r = threadIdx.x % 6;
        rb[threadIdx.x] = (e < E) ? rbf[(size_t)e * 6 + r] : 0.f;
    }
    __syncthreads();

    const int wv   = threadIdx.x >> 5;        // wave in block -> edge
    const int lane = threadIdx.x & 31;
    const int e    = ebase + wv;
    if (e >= E) return;

    const int   kb   = lane * 4;              // 4 channels per lane
    const float* re  = rb + wv * 6;
    const float4 xv  = *(const float4*)(x + (size_t)e * HIDDEN + kb);
    float* dst = agg + (size_t)idx[e] * HIDDEN + kb;

    #pragma unroll
    for (int j = 0; j < 4; ++j) {
        const float* wk = w + (kb + j) * 6;
        float s = re[0]*wk[0] + re[1]*wk[1] + re[2]*wk[2]
                + re[3]*wk[3] + re[4]*wk[4] + re[5]*wk[5];
        float h = s * ((const float*)&xv)[j];
        atomicAdd(dst + j, h);
    }
}

// ---------------------------------------------------------------------------
// WMMA f32 GEMM: out[N,O] = act(in[N,K] @ W[O,K]^T + bias)
// One wave per block computes a 16x64 output tile with 4x
// V_WMMA_F32_16X16X4_F32 accumulators; A fragment reused across 4 col tiles.
// Fragment layouts per CDNA5 ISA 7.12.2:
//   A (16x4 f32):  lane l, vgpr j  ->  M = l&15,           K = 2*(l>>4)+j
//   B (4x16 f32):  lane l, vgpr j  ->  K = 2*(l>>4)+j,     N = l&15
//   C/D (16x16):   lane l, vgpr j  ->  M = j + 8*(l>>4),   N = l&15
// K, O compile-time -> immediate-offset addressing, branch-free epilogue.
// ---------------------------------------------------------------------------
template <int K, int O, bool ACT, bool HASBIAS>
__global__ __launch_bounds__(32)
void gemm_wmma_kernel(const float* __restrict__ in,   // [N,K]
                      const float* __restrict__ W,    // [O,K] row-major
                      const float* __restrict__ bias, // [O]
                      float* __restrict__ out)        // [N,O]
{
    const int lane  = threadIdx.x;        // 0..31
    const int hi    = lane >> 4;          // 0/1
    const int lo    = lane & 15;
    const int mtile = blockIdx.y;
    const int nbase = blockIdx.x * 64;

    v8f acc[4];
    #pragma unroll
    for (int t = 0; t < 4; ++t) acc[t] = (v8f)0.0f;

    const float* arow  = in + (size_t)(mtile * 16 + lo) * K + 2 * hi;
    const float* wrow0 = W  + (size_t)(nbase + lo) * K + 2 * hi;

    #pragma unroll 4
    for (int kk = 0; kk < K; kk += 4) {
        v2f a = *(const v2f*)(arow + kk);
        #pragma unroll
        for (int t = 0; t < 4; ++t) {
            v2f b = *(const v2f*)(wrow0 + t * 16 * K + kk);
            acc[t] = __builtin_amdgcn_wmma_f32_16x16x4_f32(
                false, a, false, b, (short)0, acc[t], false, false);
        }
    }

    float* obase = out + (size_t)(mtile * 16 + 8 * hi) * O + nbase + lo;
    #pragma unroll
    for (int t = 0; t < 4; ++t) {
        float bv = 0.0f;
        if (HASBIAS) bv = bias[nbase + t * 16 + lo];
        #pragma unroll
        for (int j = 0; j < 8; ++j) {
            float v = acc[t][j];
            if (HASBIAS) v += bv;                   // skip add entirely when no bias
            if (ACT) v = v / (1.0f + __expf(-v));   // SiLU
            obase[(size_t)j * O + t * 16] = v;
        }
    }
}

// ---------------------------------------------------------------------------
// Launch: zero -> edge scatter -> up-proj -> 3x (Linear+SiLU)
// ---------------------------------------------------------------------------
extern "C" void kernel_launch(void* const* d_in, const int* in_sizes, int n_in,
                              void* d_out, int out_size, void* d_ws, size_t ws_size,
                              hipStream_t stream) {
    const float* x    = (const float*)d_in[0];
    const float* rbf  = (const float*)d_in[1];
    const int*   idx  = (const int*)  d_in[2];
    // d_in[3] = num_nodes scalar (known constant 50000)
    const float* Wrbf = (const float*)d_in[4];
    const float* Wup  = (const float*)d_in[5];
    const float* W1   = (const float*)d_in[6];
    const float* b1   = (const float*)d_in[7];
    const float* W2   = (const float*)d_in[8];
    const float* b2   = (const float*)d_in[9];
    const float* W3   = (const float*)d_in[10];
    const float* b3   = (const float*)d_in[11];
    float*       out  = (float*)d_out;

    const int E = in_sizes[2];
    const int N = NNODES;

    float* agg  = (float*)d_ws;                  // [N,128]  25.6 MB
    float* bufA = agg  + (size_t)N * HIDDEN;     // [N,256]  51.2 MB
    float* bufB = bufA + (size_t)N * OUT_EMB;    // [N,256]  51.2 MB

    // 1) zero accumulator
    {
        int n4 = (N * HIDDEN) / 4;
        zero_kernel<<<(n4 + 255) / 256, 256, 0, stream>>>(agg, n4);
    }
    // 2) edge scale + scatter-sum (8 edges per 256-thread block)
    {
        int blocks = (E + 7) / 8;
        edge_scatter_kernel<<<blocks, 256, 0, stream>>>(x, rbf, idx, Wrbf, agg, E);
    }
    // 3) node-side GEMM stack (all WMMA f32)
    dim3 grid(OUT_EMB / 64, N / 16);   // (4, 3125)
    gemm_wmma_kernel<HIDDEN,  OUT_EMB, false, false><<<grid, 32, 0, stream>>>(agg,  Wup, nullptr, bufA);
    gemm_wmma_kernel<OUT_EMB, OUT_EMB, true,  true ><<<grid, 32, 0, stream>>>(bufA, W1,  b1,      bufB);
    gemm_wmma_kernel<OUT_EMB, OUT_EMB, true,  true ><<<grid, 32, 0, stream>>>(bufB, W2,  b2,      bufA);
    gemm_wmma_kernel<OUT_EMB, OUT_EMB, true,  true ><<<grid, 32, 0, stream>>>(bufA, W3,  b3,      out);
}